// Sample_14482629722270
// MI455X (gfx1250) — compile-verified
//
#include <hip/hip_runtime.h>
#include <stdint.h>

// FPS: B=16, N=32768, C=3, M=2048. One workgroup (WGP) per batch.
#define BATCHES   16
#define NPTS      32768
#define MOUT      2048
#define THREADS   1024
#define PPT       (NPTS / THREADS)   // 32 points per thread

__device__ __forceinline__ void fps_combine(float& bv, int& bi, float ov, int oi) {
    // lexicographic max on (value, -index): matches jnp.argmax first-occurrence
    if (ov > bv || (ov == bv && oi < bi)) { bv = ov; bi = oi; }
}

__global__ __launch_bounds__(THREADS)
void fps_kernel(const float* __restrict__ pts, float* __restrict__ out) {
    const int tid  = threadIdx.x;
    const int b    = blockIdx.x;
    const int lane = tid & 31;
    const int wave = tid >> 5;

    __shared__ float s_z[NPTS];            // 128 KB: z plane resident in LDS (320 KB/WGP available)
    __shared__ float s_stage[2 * THREADS]; // 8 KB async bounce for x, y
    __shared__ float s_red[32][2];         // per-wave {value, index}
    __shared__ float s_bcast[3];           // selected {x, y, z}

    float X[PPT], Y[PPT], D[PPT];          // ~96 VGPRs + temps: no spills at 8 waves/SIMD

    // -------- Stage via CDNA5 async global->LDS loads (GVS: SGPR64 base + VGPR i32 offset) ----
    const uint64_t base  = (uint64_t)(uintptr_t)(pts + (size_t)b * 3u * NPTS);
    const unsigned lds_x = (unsigned)(uintptr_t)(&s_stage[tid]);
    const unsigned lds_y = (unsigned)(uintptr_t)(&s_stage[THREADS + tid]);

    for (int k = 0; k < PPT; ++k) {
        const int      i     = k * THREADS + tid;
        const unsigned gx    = (unsigned)((0 * NPTS + i) * 4);
        const unsigned gy    = (unsigned)((1 * NPTS + i) * 4);
        const unsigned gz    = (unsigned)((2 * NPTS + i) * 4);
        const unsigned lds_z = (unsigned)(uintptr_t)(&s_z[i]);   // z lands in its final LDS home
        asm volatile(
            "global_load_async_to_lds_b32 %0, %3, %6\n\t"
            "global_load_async_to_lds_b32 %1, %4, %6\n\t"
            "global_load_async_to_lds_b32 %2, %5, %6\n\t"
            "s_wait_asynccnt 0x0"
            :
            : "v"(lds_x), "v"(lds_y), "v"(lds_z),
              "v"(gx), "v"(gy), "v"(gz), "s"(base)
            : "memory");
        // each lane reads back only its own slots -> wave-local, no barrier needed
        X[k] = s_stage[tid];
        Y[k] = s_stage[THREADS + tid];
        D[k] = 1e10f;
        // LDS reads must retire before next iteration's async writes reuse the x/y slots
        asm volatile("s_wait_dscnt 0x0" ::: "memory");
    }

    // First selected index is 0 (CUDA FPS convention); thread 0 owns point 0 (k=0).
    if (tid == 0) {
        s_bcast[0] = X[0];
        s_bcast[1] = Y[0];
        s_bcast[2] = s_z[0];
    }
    __syncthreads();

    float* outb = out + (size_t)b * 3u * MOUT;

    for (int t = 0; t < MOUT; ++t) {
        const float qx = s_bcast[0];
        const float qy = s_bcast[1];
        const float qz = s_bcast[2];

        // Emit current selection (emit-then-update, matching the lax.scan carry order)
        if (tid == 0) {
            outb[t]            = qx;
            outb[MOUT + t]     = qy;
            outb[2 * MOUT + t] = qz;
        }

        // Update running min distance; track local argmax (first occurrence = lowest idx)
        float bv = -1.0f;
        int   bi = NPTS;
#pragma unroll
        for (int k = 0; k < PPT; ++k) {
            const int   idx = k * THREADS + tid;
            const float zk  = s_z[idx];                 // conflict-free: lanes hit consecutive banks
            const float dx  = X[k] - qx;
            const float dy  = Y[k] - qy;
            const float dz  = zk   - qz;
            const float d   = fmaf(dx, dx, fmaf(dy, dy, dz * dz));
            const float nd  = fminf(D[k], d);
            D[k] = nd;
            if (nd > bv) { bv = nd; bi = idx; }
        }

        // Stage 1: wave32 butterfly on (value, index)
#pragma unroll
        for (int m = 16; m >= 1; m >>= 1) {
            const float ov = __shfl_xor(bv, m, 32);
            const int   oi = __shfl_xor(bi, m, 32);
            fps_combine(bv, bi, ov, oi);
        }
        if (lane == 0) {
            s_red[wave][0] = bv;
            s_red[wave][1] = __int_as_float(bi);
        }
        __syncthreads();

        // Stage 2: every wave redundantly reduces the 32 candidates -> block-wide winner,
        // no extra broadcast barrier needed.
        bv = s_red[lane][0];
        bi = __float_as_int(s_red[lane][1]);
#pragma unroll
        for (int m = 16; m >= 1; m >>= 1) {
            const float ov = __shfl_xor(bv, m, 32);
            const int   oi = __shfl_xor(bi, m, 32);
            fps_combine(bv, bi, ov, oi);
        }

        // Exactly one thread in the block owns the winner's x/y registers; it publishes
        // coords via a constant-index select chain (no dynamic register indexing -> no scratch).
        if (tid == (bi & (THREADS - 1))) {
            const int ksel = bi >> 10;   // bi / THREADS
            float sx = X[0];
            float sy = Y[0];
#pragma unroll
            for (int kk = 1; kk < PPT; ++kk) {
                if (kk == ksel) { sx = X[kk]; sy = Y[kk]; }
            }
            s_bcast[0] = sx;
            s_bcast[1] = sy;
            s_bcast[2] = s_z[bi];
        }
        __syncthreads();
    }
}

extern "C" void kernel_launch(void* const* d_in, const int* in_sizes, int n_in,
                              void* d_out, int out_size, void* d_ws, size_t ws_size,
                              hipStream_t stream) {
    (void)in_sizes; (void)n_in; (void)d_ws; (void)ws_size; (void)out_size;
    const float* pts = (const float*)d_in[0];   // [16, 3, 32768] f32
    float* out = (float*)d_out;                 // [16, 3, 2048]  f32
    fps_kernel<<<BATCHES, THREADS, 0, stream>>>(pts, out);
}